// BrainGeneratorModel_76579266887905
// MI455X (gfx1250) — compile-verified
//
#include <hip/hip_runtime.h>
#include <hip/hip_bf16.h>

#define Dd 192
#define Hh 192
#define Ww 192
#define Bb 4
#define HW (Hh * Ww)                       // 36864
#define DHW ((size_t)Dd * Hh * Ww)         // 7077888
#define NVOX ((size_t)Bb * DHW)            // 28311552
#define KK 25
#define KR 12
#define KSTEPS 48                          // 192 / 4
#define NTILES 12                          // 192 / 16
#define WAVES_PER_BATCH 2304               // 192*192/16 stripes per axis
#define TOTAL_WAVES (Bb * WAVES_PER_BATCH) // 9216
// swizzled fragment buffer: per (b,axis): KSTEPS*NTILES*32 float2
#define MFRAG_PER_MAT ((size_t)KSTEPS * NTILES * 32)
#define MFRAG_TOTAL (Bb * 3 * MFRAG_PER_MAT)  // 221184 float2 = 1.77 MB

typedef float v2f __attribute__((ext_vector_type(2)));
typedef float v4f __attribute__((ext_vector_type(4)));
typedef float v8f __attribute__((ext_vector_type(8)));

#if defined(__gfx1250__) && __has_builtin(__builtin_amdgcn_wmma_f32_16x16x4_f32)
#define HAVE_WMMA_F32 1
#endif

__device__ __forceinline__ v8f wmma_acc_f32(v2f a, v2f b, v8f c) {
#ifdef HAVE_WMMA_F32
  // 8 args: (neg_a, A, neg_b, B, c_mod, C, reuse_a, reuse_b)
  return __builtin_amdgcn_wmma_f32_16x16x4_f32(false, a, false, b, (short)0, c,
                                               false, false);
#else
  #pragma unroll
  for (int i = 0; i < 8; ++i) c[i] += a.x * b.x + a.y * b.y;
  return c;
#endif
}

__device__ __forceinline__ void gauss_kernel(float sigma, float* g) {
  if (sigma >= 0.01f) {
    float s = fmaxf(sigma, 1e-3f);
    float sum = 0.0f;
    #pragma unroll
    for (int j = 0; j < KK; ++j) {
      float r = (float)(j - KR) / s;
      g[j] = __expf(-0.5f * r * r);
      sum += g[j];
    }
    float inv = 1.0f / sum;
    #pragma unroll
    for (int j = 0; j < KK; ++j) g[j] *= inv;
  } else {
    #pragma unroll
    for (int j = 0; j < KK; ++j) g[j] = 0.0f;
    g[KR] = 1.0f;  // identity operator (delta)
  }
}

// ---------------------------------------------------------------------------
// 1) Build the 192x192 blur operator (replicate padding + identity case folded
//    in) DIRECTLY in WMMA B-fragment order: element (K=ka, N=n) pair
//    (M[ka][n], M[ka+1][n]) stored as float2 at
//    idx = (((b*3+a)*KSTEPS + kstep)*NTILES + t)*32 + lane,
//    with ka = 4*kstep + 2*(lane>>4), n = t*16 + (lane&15).
//    One thread per fragment element -> perfectly coalesced b64 stores.
// ---------------------------------------------------------------------------
__global__ void build_mfrag(const float* __restrict__ sigma01,
                            float* __restrict__ Mf) {
  int tid = blockIdx.x * blockDim.x + threadIdx.x;
  if (tid >= (int)MFRAG_TOTAL) return;
  int lane  = tid & 31;
  int t     = (tid >> 5) % NTILES;
  int kstep = ((tid >> 5) / NTILES) % KSTEPS;
  int a     = ((tid >> 5) / (NTILES * KSTEPS)) % 3;
  int b     = tid / (32 * NTILES * KSTEPS * 3);

  int lm = lane & 15, lh = lane >> 4;
  int ka = 4 * kstep + 2 * lh;
  int n  = t * 16 + lm;

  float g[KK];
  gauss_kernel(sigma01[b * 3 + a] * 3.0f, g);  // MAX_SIGMA

  // M[s][n] = sum_j g[j] * (clamp(n + j - 12, 0, 191) == s)
  float v0 = 0.0f, v1 = 0.0f;
  #pragma unroll
  for (int j = 0; j < KK; ++j) {
    int s = n + j - KR;
    s = s < 0 ? 0 : (s > 191 ? 191 : s);
    if (s == ka)     v0 += g[j];
    if (s == ka + 1) v1 += g[j];
  }
  v2f* out2 = (v2f*)Mf;
  v2f val = {v0, v1};
  out2[tid] = val;
}

// ---------------------------------------------------------------------------
// 2) Fused bias-field upsample (trilinear, align_corners) + exp + multiply.
// ---------------------------------------------------------------------------
__global__ void bias_mul_kernel(const float* __restrict__ x,
                                const float* __restrict__ sb,
                                float* __restrict__ out) {
  size_t idx = (size_t)blockIdx.x * blockDim.x + threadIdx.x;
  if (idx >= NVOX) return;
  int b = (int)(idx / DHW);
  size_t rem = idx - (size_t)b * DHW;
  int d = (int)(rem / HW);
  int h = (int)((rem / Ww) % Hh);
  int w = (int)(rem % Ww);

  const float scale = 3.0f / 191.0f;  // linspace(0, n_in-1, n_out)
  float pd = d * scale, ph = h * scale, pw = w * scale;
  int id = (int)pd; id = id > 2 ? 2 : id;
  int ih = (int)ph; ih = ih > 2 ? 2 : ih;
  int iw = (int)pw; iw = iw > 2 ? 2 : iw;
  float fd = pd - (float)id, fh = ph - (float)ih, fw = pw - (float)iw;
  float wd[2] = {1.0f - fd, fd};
  float wh[2] = {1.0f - fh, fh};
  float wwv[2] = {1.0f - fw, fw};

  const float* s = sb + b * 64;  // 4*4*4 per batch
  float bias = 0.0f;
  #pragma unroll
  for (int cd = 0; cd < 2; ++cd)
    #pragma unroll
    for (int ch = 0; ch < 2; ++ch)
      #pragma unroll
      for (int cw = 0; cw < 2; ++cw)
        bias += wd[cd] * wh[ch] * wwv[cw] *
                s[((id + cd) * 4 + (ih + ch)) * 4 + (iw + cw)];

  out[idx] = x[idx] * __expf(bias * 0.7f);  // BIAS_STD
}

// ---------------------------------------------------------------------------
// 3) Axis blur as WMMA GEMM: Out[row, t] = sum_s In[row, s] * M[s, t].
//    One wave owns a 16-row x 192-col output stripe (12 C tiles, 96 VGPRs);
//    K swept in steps of 4 via V_WMMA_F32_16X16X4_F32; B fragments come
//    pre-swizzled (one coalesced b64 load per tile per k-step).
//    AXIS: 0=D (stride 36864), 1=H (stride 192), 2=W (stride 1).
// ---------------------------------------------------------------------------
template <int AXIS>
__global__ void __launch_bounds__(128) blur_gemm(const float* __restrict__ in,
                                                 float* __restrict__ out,
                                                 const float* __restrict__ Mb) {
  int wave = (int)((blockIdx.x * blockDim.x + threadIdx.x) >> 5);
  int lane = threadIdx.x & 31;
  int b = wave / WAVES_PER_BATCH;
  int r = wave % WAVES_PER_BATCH;

  size_t base;
  size_t axis_stride;
  int row0;
  if (AXIS == 0) {        // blur along D; rows = contiguous w within (b,h) slab
    int h = r / NTILES;
    row0 = (r % NTILES) * 16;
    base = (size_t)b * DHW + (size_t)h * Ww;
    axis_stride = HW;
  } else if (AXIS == 1) { // blur along H; rows = contiguous w within (b,d) slab
    int d = r / NTILES;
    row0 = (r % NTILES) * 16;
    base = (size_t)b * DHW + (size_t)d * HW;
    axis_stride = Ww;
  } else {                // blur along W; rows = (d,h) lines, axis contiguous
    row0 = r * 16;
    base = (size_t)b * DHW;
    axis_stride = 1;
  }
  const size_t row_stride = (AXIS == 2) ? (size_t)Ww : (size_t)1;

  const v2f* Mf =
      (const v2f*)Mb + (size_t)(b * 3 + AXIS) * MFRAG_PER_MAT;
  int lm = lane & 15;  // A row / C col within tile
  int lh = lane >> 4;  // half-wave select

  const float* Ap = in + base + row_stride * (size_t)(row0 + lm);

  v8f acc[NTILES] = {};
  for (int kstep = 0; kstep < KSTEPS; ++kstep) {
    int ka = 4 * kstep + 2 * lh;  // A 16x4 f32: VGPR0 K={0|2}, VGPR1 K={1|3}
    v2f a;
    if (AXIS == 2) {
      a = *(const v2f*)(Ap + ka);  // contiguous pair -> single b64 load
    } else {
      a.x = Ap[axis_stride * (size_t)ka];
      a.y = Ap[axis_stride * (size_t)(ka + 1)];
    }
    const v2f* Mrow = Mf + (size_t)kstep * (NTILES * 32) + lane;
    #pragma unroll
    for (int t = 0; t < NTILES; ++t) {
      acc[t] = wmma_acc_f32(a, Mrow[t * 32], acc[t]);  // coalesced b64 load
    }
  }

  #pragma unroll
  for (int t = 0; t < NTILES; ++t) {
    if (AXIS == 2) {
      // n contiguous across lanes -> coalesced b32 stores
      #pragma unroll
      for (int v = 0; v < 8; ++v) {
        int m = row0 + v + 8 * lh;  // C/D: VGPR v -> M = v + 8*(lane/16)
        int n = t * 16 + lm;        // N = lane%16
        out[base + row_stride * (size_t)m + (size_t)n] = acc[t][v];
      }
    } else {
      // per-lane 8 contiguous floats (m stride 1) -> two b128 stores
      size_t addr =
          base + (size_t)(row0 + 8 * lh) + axis_stride * (size_t)(t * 16 + lm);
      v4f lo = {acc[t][0], acc[t][1], acc[t][2], acc[t][3]};
      v4f hi = {acc[t][4], acc[t][5], acc[t][6], acc[t][7]};
      *(v4f*)(out + addr)     = lo;
      *(v4f*)(out + addr + 4) = hi;
    }
  }
}

// ---------------------------------------------------------------------------
// 4) Label remap LUT + gather.
// ---------------------------------------------------------------------------
__global__ void build_table(const int* __restrict__ src,
                            const int* __restrict__ dst,
                            int* __restrict__ table) {
  int t = threadIdx.x;  // 128 threads
  int m = 0;
  for (int i = 0; i < 32; ++i)
    if (src[i] == t) m = dst[i];
  table[t] = m;
}

__global__ void remap_labels(const int* __restrict__ labels,
                             const int* __restrict__ table,
                             int* __restrict__ out) {
  size_t idx = (size_t)blockIdx.x * blockDim.x + threadIdx.x;
  if (idx >= NVOX) return;
  out[idx] = table[labels[idx] & 127];
}

// ---------------------------------------------------------------------------
extern "C" void kernel_launch(void* const* d_in, const int* in_sizes, int n_in,
                              void* d_out, int out_size, void* d_ws,
                              size_t ws_size, hipStream_t stream) {
  const float* x      = (const float*)d_in[0];
  const float* sb     = (const float*)d_in[1];
  const float* sig    = (const float*)d_in[2];
  const int*   labels = (const int*)d_in[3];
  const int*   srcv   = (const int*)d_in[4];
  const int*   dstv   = (const int*)d_in[5];

  float* img     = (float*)d_out;          // output 0: blurred image
  float* scratch = img + NVOX;             // labels half doubles as f32 scratch
  int*   lab_out = (int*)d_out + NVOX;     // output 1: remapped labels (final)

  float* Mf    = (float*)d_ws;                          // fragments (1.77 MB)
  int*   table = (int*)((float*)d_ws + 2 * MFRAG_TOTAL); // 128 ints after

  // blur operators, pre-swizzled to WMMA fragment order (tiny)
  build_mfrag<<<(int)((MFRAG_TOTAL + 255) / 256), 256, 0, stream>>>(sig, Mf);
  // bias * exp fused, into scratch (labels half of d_out)
  bias_mul_kernel<<<(int)((NVOX + 255) / 256), 256, 0, stream>>>(x, sb, scratch);
  // separable blur via WMMA GEMM, ping-pong; order D,H,W (operators commute)
  blur_gemm<0><<<TOTAL_WAVES / 4, 128, 0, stream>>>(scratch, img, Mf);
  blur_gemm<1><<<TOTAL_WAVES / 4, 128, 0, stream>>>(img, scratch, Mf);
  blur_gemm<2><<<TOTAL_WAVES / 4, 128, 0, stream>>>(scratch, img, Mf);
  // labels last (overwrites the scratch half)
  build_table<<<1, 128, 0, stream>>>(srcv, dstv, table);
  remap_labels<<<(int)((NVOX + 255) / 256), 256, 0, stream>>>(labels, table,
                                                              lab_out);
}